// DepthScaleCorrector_32744830665233
// MI455X (gfx1250) — compile-verified
//
#include <hip/hip_runtime.h>
#include <hip/hip_bf16.h>

// ---------------------------------------------------------------------------
// DepthScaleCorrector for MI455X (gfx1250, wave32)
//   in0: non_scale_dense (16,4,768,1024) f32
//   in1: sparse_depth    (16,1,768,1024) f32
//   out: [corrected 16*4*768*1024][best_corrected 16*768*1024][best_idx 16]
// ---------------------------------------------------------------------------

typedef __attribute__((ext_vector_type(2))) float v2f;
typedef __attribute__((ext_vector_type(8))) float v8f;

#define NB 16            // batches
#define NC 4             // channels
#define HW 786432        // 768*1024
#define GPB 196608       // HW/4 float4 groups per batch-channel
#define BLKS 64          // blocks per batch in streaming kernels
#define GPBLK 3072       // GPB / BLKS
#define ITERS 12         // GPBLK / 256

// workspace layout (float indices)
#define WS0       0        // stats partials: 16 batches * 64 blocks * 16 q
#define WS1       16384    // err partials:   16 batches * 64 blocks * 16 q
#define WS_SCALE  32768    // [16][4]
#define WS_BIAS   32832    // [16][4]
#define WS_N      32896    // [16]
#define WS_BESTI  32912    // [16] (as int*)

#define OUT_BEST  50331648   // float offset of best_corrected
#define OUT_IDX   62914560   // float offset of best_idx

// ---------------------------------------------------------------------------
// Reduce 64 partial records of 16 quantities into smem[0..15] using chained
// V_WMMA_F32_16X16X4_F32:  D(16x16) += A(16x4 partials) x B(4x16 ones).
// Row m of A = quantity m; K dimension walks the 64 block-partials, 4 at a
// time, 16 accumulating WMMAs. All D columns are identical; per ISA layout
// lane0 holds M=0..7 in VGPR0..7 and lane16 holds M=8..15.
// Must be called by one full wave32 with uniform control flow (EXEC all 1s).
// ---------------------------------------------------------------------------
__device__ inline void wmma_reduce64(const float* __restrict__ base, int lane,
                                     float* __restrict__ smem) {
#if defined(__gfx1250__) && __has_builtin(__builtin_amdgcn_wmma_f32_16x16x4_f32)
  v8f acc = {0.f, 0.f, 0.f, 0.f, 0.f, 0.f, 0.f, 0.f};
  v2f bones;
  bones.x = 1.0f;
  bones.y = 1.0f;
  const int m  = lane & 15;              // A-matrix row = quantity index
  const int kb = (lane < 16) ? 0 : 2;    // lanes 16-31 carry K=2,3 (ISA 7.12.2)
#pragma unroll
  for (int j = 0; j < 16; ++j) {         // 16 chunks of K=4 -> 64 blocks
    v2f a;
    a.x = base[(4 * j + kb + 0) * 16 + m];
    a.y = base[(4 * j + kb + 1) * 16 + m];
    acc = __builtin_amdgcn_wmma_f32_16x16x4_f32(
        /*neg_a=*/false, a, /*neg_b=*/false, bones,
        /*c_mod=*/(short)0, acc, /*reuse_a=*/false, /*reuse_b=*/false);
  }
  if (lane == 0) {
#pragma unroll
    for (int r = 0; r < 8; ++r) smem[r] = acc[r];        // M = 0..7
  }
  if (lane == 16) {
#pragma unroll
    for (int r = 0; r < 8; ++r) smem[8 + r] = acc[r];    // M = 8..15
  }
#else
  if (lane < 16) {
    float s = 0.0f;
    for (int blk = 0; blk < 64; ++blk) s += base[blk * 16 + lane];
    smem[lane] = s;
  }
#endif
  __syncthreads();
}

// block-level reduction of 16 per-thread quantities -> one 16-float record
__device__ inline void block_reduce16(float (&p)[16], int tid, int b, int blk,
                                      float* __restrict__ dst) {
  __shared__ float lds[8 * 16];
#pragma unroll
  for (int q = 0; q < 16; ++q) {
    float v = p[q];
    v += __shfl_xor(v, 16, 32);
    v += __shfl_xor(v, 8, 32);
    v += __shfl_xor(v, 4, 32);
    v += __shfl_xor(v, 2, 32);
    v += __shfl_xor(v, 1, 32);
    p[q] = v;
  }
  const int wid = tid >> 5, lane = tid & 31;
  if (lane == 0) {
#pragma unroll
    for (int q = 0; q < 16; ++q) lds[wid * 16 + q] = p[q];
  }
  __syncthreads();
  if (tid < 16) {
    float s = 0.0f;
#pragma unroll
    for (int w = 0; w < 8; ++w) s += lds[w * 16 + tid];
    dst[(size_t)(b * BLKS + blk) * 16 + tid] = s;
  }
}

// ---------------------------------------------------------------------------
// Pass 1: masked sums. Quantities: q=c*3+{0,1,2} = {Σx, Σx², Σxy}; q12=Σy;
// q13=Σm; q14,q15=0 (padding so the WMMA A-tile is 16 rows).
// ---------------------------------------------------------------------------
__global__ void __launch_bounds__(256) k_stats(const float4* __restrict__ nsd4,
                                               const float4* __restrict__ sp4,
                                               float* __restrict__ ws0) {
  const int b = blockIdx.x >> 6;
  const int blk = blockIdx.x & 63;
  const int tid = threadIdx.x;
  float p[16];
#pragma unroll
  for (int q = 0; q < 16; ++q) p[q] = 0.0f;

  const float4* spb = sp4 + (size_t)b * GPB;
  const int gbase = blk * GPBLK + tid;
#pragma unroll 2
  for (int i = 0; i < ITERS; ++i) {
    const int g = gbase + i * 256;
    const float4 s = spb[g];
    const float m0 = (s.x > 1e-6f && s.x <= 20.0f) ? 1.0f : 0.0f;
    const float m1 = (s.y > 1e-6f && s.y <= 20.0f) ? 1.0f : 0.0f;
    const float m2 = (s.z > 1e-6f && s.z <= 20.0f) ? 1.0f : 0.0f;
    const float m3 = (s.w > 1e-6f && s.w <= 20.0f) ? 1.0f : 0.0f;
    const float sm0 = s.x * m0, sm1 = s.y * m1, sm2 = s.z * m2, sm3 = s.w * m3;
    p[12] += (sm0 + sm1) + (sm2 + sm3);
    p[13] += (m0 + m1) + (m2 + m3);
#pragma unroll
    for (int c = 0; c < 4; ++c) {
      const float4 x = nsd4[(size_t)(b * NC + c) * GPB + g];
      const float a0 = x.x * m0, a1 = x.y * m1, a2 = x.z * m2, a3 = x.w * m3;
      p[c * 3 + 0] += (a0 + a1) + (a2 + a3);
      p[c * 3 + 1] += (x.x * a0 + x.y * a1) + (x.z * a2 + x.w * a3);
      p[c * 3 + 2] += (sm0 * x.x + sm1 * x.y) + (sm2 * x.z + sm3 * x.w);
    }
  }
  block_reduce16(p, tid, b, blk, ws0);
}

// ---------------------------------------------------------------------------
// Pass 2: one wave32 per batch. WMMA-reduce the 64 partial records, then
// lanes 0..3 solve the 2x2 normal equations per channel.
// ---------------------------------------------------------------------------
__global__ void __launch_bounds__(32) k_solve(float* __restrict__ ws) {
  const int b = blockIdx.x;
  const int lane = threadIdx.x;
  __shared__ float smem[16];
  wmma_reduce64(ws + WS0 + (size_t)b * (BLKS * 16), lane, smem);
  if (lane < 4) {
    const int c = lane;
    const float n = smem[13];
    const float xs = smem[c * 3 + 0];
    const float x2 = smem[c * 3 + 1];
    const float xy = smem[c * 3 + 2];
    const float ys = smem[12];
    const float det = n * x2 - xs * xs;
    const bool ok = (n >= 10.0f) && (fabsf(det) >= 1e-8f);
    const float sd = ok ? det : 1.0f;
    const float sc = ok ? (n * xy - xs * ys) / sd : 1.0f;
    const float bi = ok ? (x2 * ys - xs * xy) / sd : 0.0f;
    ws[WS_SCALE + b * 4 + c] = sc;
    ws[WS_BIAS + b * 4 + c] = bi;
    if (c == 0) ws[WS_N + b] = n;
  }
}

// ---------------------------------------------------------------------------
// Pass 3: corrected = clip(scale*x + bias, 0, 20) -> d_out, plus masked
// squared-error partial sums (q=0..3, padded to 16) -> ws1.
// ---------------------------------------------------------------------------
__global__ void __launch_bounds__(256) k_correct(const float4* __restrict__ nsd4,
                                                 const float4* __restrict__ sp4,
                                                 float4* __restrict__ out4,
                                                 float* __restrict__ ws) {
  const int b = blockIdx.x >> 6;
  const int blk = blockIdx.x & 63;
  const int tid = threadIdx.x;
  float sc[4], bi[4];
#pragma unroll
  for (int c = 0; c < 4; ++c) {
    sc[c] = ws[WS_SCALE + b * 4 + c];
    bi[c] = ws[WS_BIAS + b * 4 + c];
  }
  float p[16];
#pragma unroll
  for (int q = 0; q < 16; ++q) p[q] = 0.0f;

  const float4* spb = sp4 + (size_t)b * GPB;
  const int gbase = blk * GPBLK + tid;
#pragma unroll 2
  for (int i = 0; i < ITERS; ++i) {
    const int g = gbase + i * 256;
    const float4 s = spb[g];
    const float m0 = (s.x > 1e-6f && s.x <= 20.0f) ? 1.0f : 0.0f;
    const float m1 = (s.y > 1e-6f && s.y <= 20.0f) ? 1.0f : 0.0f;
    const float m2 = (s.z > 1e-6f && s.z <= 20.0f) ? 1.0f : 0.0f;
    const float m3 = (s.w > 1e-6f && s.w <= 20.0f) ? 1.0f : 0.0f;
#pragma unroll
    for (int c = 0; c < 4; ++c) {
      const size_t idx = (size_t)(b * NC + c) * GPB + g;
      const float4 x = nsd4[idx];
      float4 r;
      r.x = fminf(fmaxf(fmaf(sc[c], x.x, bi[c]), 0.0f), 20.0f);
      r.y = fminf(fmaxf(fmaf(sc[c], x.y, bi[c]), 0.0f), 20.0f);
      r.z = fminf(fmaxf(fmaf(sc[c], x.z, bi[c]), 0.0f), 20.0f);
      r.w = fminf(fmaxf(fmaf(sc[c], x.w, bi[c]), 0.0f), 20.0f);
      out4[idx] = r;
      const float d0 = r.x - s.x, d1 = r.y - s.y, d2 = r.z - s.z, d3 = r.w - s.w;
      p[c] += (d0 * d0 * m0 + d1 * d1 * m1) + (d2 * d2 * m2 + d3 * d3 * m3);
    }
  }
  block_reduce16(p, tid, b, blk, ws + WS1);
}

// ---------------------------------------------------------------------------
// Pass 4: WMMA-reduce error partials, MSE + first-min argmin per batch.
// ---------------------------------------------------------------------------
__global__ void __launch_bounds__(32) k_select(float* __restrict__ ws,
                                               float* __restrict__ out) {
  const int b = blockIdx.x;
  const int lane = threadIdx.x;
  __shared__ float smem[16];
  wmma_reduce64(ws + WS1 + (size_t)b * (BLKS * 16), lane, smem);
  if (lane == 0) {
    const float n = ws[WS_N + b];
    const float denom = fmaxf(n, 1.0f);
    int best = 0;
    float bm = (n > 0.0f) ? smem[0] / denom : __builtin_inff();
#pragma unroll
    for (int c = 1; c < 4; ++c) {
      const float mse = (n > 0.0f) ? smem[c] / denom : __builtin_inff();
      if (mse < bm) { bm = mse; best = c; }
    }
    ((int*)ws)[WS_BESTI + b] = best;
    out[OUT_IDX + b] = (float)best;
  }
}

// ---------------------------------------------------------------------------
// Pass 5: gather best channel into best_corrected.
// ---------------------------------------------------------------------------
__global__ void __launch_bounds__(256) k_gather(const float4* __restrict__ corr4,
                                                float4* __restrict__ best4,
                                                const int* __restrict__ bidx) {
  const int b = blockIdx.x >> 6;
  const int blk = blockIdx.x & 63;
  const int tid = threadIdx.x;
  const int best = bidx[b];
  const size_t srcb = (size_t)(b * NC + best) * GPB;
  const size_t dstb = (size_t)b * GPB;
  const int gbase = blk * GPBLK + tid;
#pragma unroll
  for (int i = 0; i < ITERS; ++i) {
    const int g = gbase + i * 256;
    best4[dstb + g] = corr4[srcb + g];
  }
}

extern "C" void kernel_launch(void* const* d_in, const int* in_sizes, int n_in,
                              void* d_out, int out_size, void* d_ws, size_t ws_size,
                              hipStream_t stream) {
  const float* nsd = (const float*)d_in[0];
  const float* sp = (const float*)d_in[1];
  float* out = (float*)d_out;
  float* ws = (float*)d_ws;
  (void)in_sizes; (void)n_in; (void)out_size; (void)ws_size;

  k_stats<<<dim3(NB * BLKS), dim3(256), 0, stream>>>(
      (const float4*)nsd, (const float4*)sp, ws + WS0);
  k_solve<<<dim3(NB), dim3(32), 0, stream>>>(ws);
  k_correct<<<dim3(NB * BLKS), dim3(256), 0, stream>>>(
      (const float4*)nsd, (const float4*)sp, (float4*)out, ws);
  k_select<<<dim3(NB), dim3(32), 0, stream>>>(ws, out);
  k_gather<<<dim3(NB * BLKS), dim3(256), 0, stream>>>(
      (const float4*)out, (float4*)(out + OUT_BEST), (const int*)ws + WS_BESTI);
}